// GCN_63187558859328
// MI455X (gfx1250) — compile-verified
//
#include <hip/hip_runtime.h>

typedef float v2f __attribute__((ext_vector_type(2)));
typedef float v8f __attribute__((ext_vector_type(8)));

#define DIN  512
#define DHID 512
#define DOUT 256

// ---------------- elementwise helpers ----------------

__global__ void k_fill(float* __restrict__ p, float val, int n) {
  int i = blockIdx.x * blockDim.x + threadIdx.x;
  if (i < n) p[i] = val;
}

__global__ void k_zero4(float4* __restrict__ p, int n4) {
  int i = blockIdx.x * blockDim.x + threadIdx.x;
  if (i < n4) p[i] = make_float4(0.f, 0.f, 0.f, 0.f);
}

__device__ __forceinline__ void atomAddF(float* p, float v) {
  // relaxed agent-scope fp32 add -> global_atomic_add_f32 (non-returning)
  __hip_atomic_fetch_add(p, v, __ATOMIC_RELAXED, __HIP_MEMORY_SCOPE_AGENT);
}

// ---------------- degree / normalization ----------------

__global__ void k_deg_count(const int* __restrict__ tgt, float* __restrict__ deg, int E) {
  int e = blockIdx.x * blockDim.x + threadIdx.x;
  if (e < E) atomAddF(&deg[tgt[e]], 1.0f);
}

__global__ void k_rsqrt(float* __restrict__ p, int n) {
  int i = blockIdx.x * blockDim.x + threadIdx.x;
  if (i < n) {
    float d = p[i];
    p[i] = (d > 0.f) ? rsqrtf(d) : 0.f;
  }
}

// ---------------- fp32 GEMM via V_WMMA_F32_16X16X4_F32 ----------------
// C[M,N] = A[M,K] @ B[K,N].  K, N compile-time (strides fold into the 24-bit
// immediate-offset field of global_load); M runtime, multiple of 16.
// Each wave: 64(M) x 32(N) tile = 4 M-subtiles x 2 N-tiles, 8 accumulators;
// per k-step: 4 x b64 A loads + 4 x b32 B loads -> 8 WMMAs.
// The M-tail decision (clamped duplicate subtiles) is hoisted OUT of the
// k-loop via the TAIL template parameter so the hot loop has no branches.
// Fragment layouts per CDNA5 ISA 7.12.2:
//   A 16x4 : lane (m = lane&15, g = lane>>4), vgpr v -> A[m][k0 + v + 2g]
//   B 4x16 : lane n = lane&15, g,             vgpr v -> B[k0 + v + 2g][n0+n]
//   C 16x16: lane n, g,                       vgpr v -> C[row0 + v + 8g][n0+n]

template <int K, int N, bool SQUARE_A, bool TAIL>
__device__ __forceinline__ void gemm_kloop(const float* __restrict__ a0,
                                           const float* __restrict__ a1,
                                           const float* __restrict__ a2,
                                           const float* __restrict__ a3,
                                           const float* __restrict__ bcol,
                                           v8f acc[4][2]) {
  for (int k0 = 0; k0 < K; k0 += 4) {
    v2f a[4];
    if (!TAIL) {
      a[0] = *(const v2f*)(a0 + k0);
      a[1] = *(const v2f*)(a0 + k0 + 16 * K);   // immediate offsets off one base
      a[2] = *(const v2f*)(a0 + k0 + 32 * K);
      a[3] = *(const v2f*)(a0 + k0 + 48 * K);
    } else {
      a[0] = *(const v2f*)(a0 + k0);
      a[1] = *(const v2f*)(a1 + k0);
      a[2] = *(const v2f*)(a2 + k0);
      a[3] = *(const v2f*)(a3 + k0);
    }
    if (SQUARE_A) {
#pragma unroll
      for (int i = 0; i < 4; ++i) { a[i].x *= a[i].x; a[i].y *= a[i].y; }
    }
    const float* bp = bcol + (size_t)k0 * N;
    v2f b0, b1;
    b0.x = bp[0];  b0.y = bp[N + 0];            // imm offsets (N compile-time)
    b1.x = bp[16]; b1.y = bp[N + 16];
#pragma unroll
    for (int i = 0; i < 4; ++i) {
      acc[i][0] = __builtin_amdgcn_wmma_f32_16x16x4_f32(false, a[i], false, b0,
                                                        (short)0, acc[i][0], false, false);
      acc[i][1] = __builtin_amdgcn_wmma_f32_16x16x4_f32(false, a[i], false, b1,
                                                        (short)0, acc[i][1], false, false);
    }
  }
}

template <int K, int N, bool SQUARE_A>
__global__ __launch_bounds__(256)
void k_gemm_wmma(const float* __restrict__ A, const float* __restrict__ B,
                 float* __restrict__ C, int M) {
  const int lane = threadIdx.x & 31;
  const int wid  = threadIdx.x >> 5;
  const int m = lane & 15;
  const int g = lane >> 4;

  const int row0 = (blockIdx.x * 8 + wid) * 64;  // wave-uniform
  if (row0 >= M) return;                         // EXEC stays all-1 on live path
  const int col0 = blockIdx.y * 32;

  int  r[4];
  bool st[4];
#pragma unroll
  for (int i = 0; i < 4; ++i) {
    int rr = row0 + 16 * i;
    st[i] = (rr < M);                  // M is a multiple of 16
    r[i]  = st[i] ? rr : (M - 16);     // clamp: valid loads, discarded result
  }

  v8f acc[4][2] = {};
  const float* a0   = A + (size_t)(r[0] + m) * K + 2 * g;
  const float* bcol = B + (size_t)(2 * g) * N + col0 + m;  // m = (lane&15) is N-index

  if (row0 + 64 <= M) {
    gemm_kloop<K, N, SQUARE_A, false>(a0, a0, a0, a0, bcol, acc);
  } else {
    const float* a1 = A + (size_t)(r[1] + m) * K + 2 * g;
    const float* a2 = A + (size_t)(r[2] + m) * K + 2 * g;
    const float* a3 = A + (size_t)(r[3] + m) * K + 2 * g;
    gemm_kloop<K, N, SQUARE_A, true>(a0, a1, a2, a3, bcol, acc);
  }

#pragma unroll
  for (int i = 0; i < 4; ++i) {
    if (!st[i]) continue;              // wave-uniform
    float* crow = C + (size_t)(r[i] + 8 * g) * N + col0 + m;
#pragma unroll
    for (int v = 0; v < 8; ++v) {
      crow[(size_t)v * N]      = acc[i][0][v];
      crow[(size_t)v * N + 16] = acc[i][1][v];
    }
  }
}

// ---------------- edge scatter:  out[tgt] += dinv[src]*dinv[tgt] * feat[src] ----
// One wave per (edge or self-loop).  D is 512 or 256 (multiple of 128).
__global__ __launch_bounds__(256)
void k_scatter(const float* __restrict__ feat, const float* __restrict__ dinv,
               const int* __restrict__ src, const int* __restrict__ tgt,
               float* __restrict__ out, int E, int N, int D) {
  int wave = (int)((blockIdx.x * blockDim.x + threadIdx.x) >> 5);
  int lane = threadIdx.x & 31;
  int total = E + N;
  if (wave >= total) return;

  int s, t;
  if (wave < E) { s = src[wave]; t = tgt[wave]; }
  else          { s = t = wave - E; }          // self-loop

  float norm = dinv[s] * dinv[t];
  const float4* fs = (const float4*)(feat + (size_t)s * D);
  float* od = out + (size_t)t * D;
  int nvec = D >> 2;
  for (int v = lane; v < nvec; v += 32) {
    float4 val = fs[v];
    int base = v * 4;
    atomAddF(od + base + 0, norm * val.x);
    atomAddF(od + base + 1, norm * val.y);
    atomAddF(od + base + 2, norm * val.z);
    atomAddF(od + base + 3, norm * val.w);
  }
}

// ---------------- driver ----------------

extern "C" void kernel_launch(void* const* d_in, const int* in_sizes, int n_in,
                              void* d_out, int out_size, void* d_ws, size_t ws_size,
                              hipStream_t stream) {
  const float* x  = (const float*)d_in[0];
  const int*   ei = (const int*)d_in[1];
  const float* W1 = (const float*)d_in[2];
  const float* W2 = (const float*)d_in[3];
  float* out = (float*)d_out;

  const int N = in_sizes[0] / DIN;     // 10000
  const int E = in_sizes[1] / 2;       // 160000
  const int* src = ei;                 // edge_index[0]
  const int* tgt = ei + E;             // edge_index[1]

  char* ws = (char*)d_ws;
  size_t off = 0;
  float* dinv = (float*)(ws + off); off += ((size_t)N * 4 + 255) & ~(size_t)255;
  float* xw   = (float*)(ws + off); off += ((size_t)N * DHID * 4 + 255) & ~(size_t)255;
  float* h    = (float*)(ws + off); off += ((size_t)N * DHID * 4 + 255) & ~(size_t)255;
  float* hw   = xw;                    // xw dead after layer-1 scatter; reuse (DOUT <= DHID)
  (void)ws_size; (void)n_in; (void)out_size;

  const int nEdges = E + N;
  const int mBlocks = (N + 511) / 512; // 8 waves/block * 64 rows/wave

  // degrees (incl. self-loop) -> dinv = deg^{-1/2}
  k_fill<<<(N + 255) / 256, 256, 0, stream>>>(dinv, 1.0f, N);
  k_deg_count<<<(E + 255) / 256, 256, 0, stream>>>(tgt, dinv, E);
  k_rsqrt<<<(N + 255) / 256, 256, 0, stream>>>(dinv, N);

  // layer 1: xw = x @ W1
  {
    dim3 grid(mBlocks, DHID / 32);
    k_gemm_wmma<DIN, DHID, false><<<grid, 256, 0, stream>>>(x, W1, xw, N);
  }
  // h = scatter(norm * xw[src] -> tgt)
  {
    int n4 = N * DHID / 4;
    k_zero4<<<(n4 + 255) / 256, 256, 0, stream>>>((float4*)h, n4);
    k_scatter<<<(nEdges + 7) / 8, 256, 0, stream>>>(xw, dinv, src, tgt, h, E, N, DHID);
  }
  // layer 2: hw = (h^2) @ W2   (square fused into A-fragment load)
  {
    dim3 grid(mBlocks, DOUT / 32);
    k_gemm_wmma<DHID, DOUT, true><<<grid, 256, 0, stream>>>(h, W2, hw, N);
  }
  // out = scatter(norm * hw[src] -> tgt)
  {
    int o4 = N * DOUT / 4;
    k_zero4<<<(o4 + 255) / 256, 256, 0, stream>>>((float4*)out, o4);
    k_scatter<<<(nEdges + 7) / 8, 256, 0, stream>>>(hw, dinv, src, tgt, out, E, N, DOUT);
  }
}